// LinearMPO_27917287424190
// MI455X (gfx1250) — compile-verified
//
#include <hip/hip_runtime.h>

// Problem dims (match reference): L=32 sites, D=4 MPO bond, d=2 phys, CHI=256 MPS bond.
constexpr int L_   = 32;
constexpr int D_   = 4;
constexpr int d_   = 2;
constexpr int CHI_ = 256;
constexpr int BOND = D_ * CHI_;   // 1024 fused bond

typedef __attribute__((ext_vector_type(2))) float v2f;
typedef __attribute__((ext_vector_type(8))) float v8f;

// out[l, a, x, i, b, y] = sum_j cores[l,a,i,j,b] * mps[l,x,j,y]
// Per site l: GEMM A[32x2] x B[2x65536] done as V_WMMA_F32_16X16X4_F32 tiles
// (M split into 2 tiles of 16, K=2 padded to 4 with zeros, N=16 y-columns).
__global__ __launch_bounds__(256) void mpo_mps_wmma(
    const float* __restrict__ mps,    // [L, CHI, d, CHI]
    const float* __restrict__ cores,  // [L, D, d, d, D]
    float* __restrict__ out)          // [L, D*CHI, d, D*CHI]
{
  const int blk  = blockIdx.x;      // (l, x)
  const int l    = blk >> 8;
  const int x    = blk & 255;
  const int tid  = threadIdx.x;
  const int wave = tid >> 5;        // 8 waves per block
  const int lane = tid & 31;
  const int n    = lane & 15;       // N column within tile / M row for A setup
  const bool hi  = lane >= 16;      // lanes 16-31: K=2,3 half (zero-padded K)

  // ---- A matrices: 16x4 f32, rows m = aLocal*8 + i*4 + b, K=2 real ----
  // Lane p<16: v0 = A[m, K=0], v1 = A[m, K=1]. Lane p>=16: K=2,3 -> 0.
  v2f A0 = {0.f, 0.f};              // atile 0: a in {0,1}
  v2f A1 = {0.f, 0.f};              // atile 1: a in {2,3}
  if (!hi) {
    const int m  = n;
    const int aL = m >> 3;
    const int i  = (m >> 2) & 1;
    const int b  = m & 3;
    const float* cb = cores + l * (D_ * d_ * d_ * D_);   // 64 floats per site
    const int base0 = (aL)     * 16 + i * 8 + b;         // + j*4 for K index
    const int base1 = (2 + aL) * 16 + i * 8 + b;
    A0.x = cb[base0 + 0];  A0.y = cb[base0 + 4];
    A1.x = cb[base1 + 0];  A1.y = cb[base1 + 4];
  }

  const float* mrow0 = mps + ((l * CHI_ + x) * d_ + 0) * CHI_;  // j=0 row
  const float* mrow1 = mrow0 + CHI_;                            // j=1 row

  // Each wave covers 2 tiles of 16 y-columns (16 tiles total per (l,x)).
  for (int yt = wave * 2; yt < wave * 2 + 2; ++yt) {
    const int y0 = yt * 16;

    // ---- B matrix: 4x16 f32, lanes 0-15 = K rows 0/1, lanes 16-31 zero ----
    v2f B = {0.f, 0.f};
    if (!hi) {
      B.x = mrow0[y0 + n];
      B.y = mrow1[y0 + n];
    }

    v8f C0 = {};
    v8f C1 = {};
    // 8 args: (neg_a, A, neg_b, B, c_mod, C, reuse_a, reuse_b) — EXEC all-1s here.
    C0 = __builtin_amdgcn_wmma_f32_16x16x4_f32(false, A0, false, B, (short)0, C0,
                                               false, false);
    C1 = __builtin_amdgcn_wmma_f32_16x16x4_f32(false, A1, false, B, (short)0, C1,
                                               false, false);

    // ---- Store D: VGPR r holds (M=r, lanes 0-15) and (M=r+8, lanes 16-31) ----
    // out flat idx = ((l*1024 + a*256 + x)*2 + i)*1024 + b*256 + y
#pragma unroll
    for (int r = 0; r < 8; ++r) {
      const int m  = r + (hi ? 8 : 0);
      const int aL = m >> 3;
      const int i  = (m >> 2) & 1;
      const int b  = m & 3;
      const int y  = y0 + n;
      const int idx0 = ((l * BOND + (aL)     * CHI_ + x) * d_ + i) * BOND + b * CHI_ + y;
      const int idx1 = ((l * BOND + (2 + aL) * CHI_ + x) * d_ + i) * BOND + b * CHI_ + y;
      out[idx0] = C0[r];
      out[idx1] = C1[r];
    }
  }
}

extern "C" void kernel_launch(void* const* d_in, const int* in_sizes, int n_in,
                              void* d_out, int out_size, void* d_ws, size_t ws_size,
                              hipStream_t stream) {
  const float* mps   = (const float*)d_in[0];  // [32,256,2,256] f32
  const float* cores = (const float*)d_in[1];  // [32,4,2,2,4]  f32
  float* out = (float*)d_out;                  // [32,1024,2,1024] f32

  dim3 grid(L_ * CHI_);   // 8192 blocks: one per (l, x)
  dim3 block(256);        // 8 wave32s
  mpo_mps_wmma<<<grid, block, 0, stream>>>(mps, cores, out);
}